// CorrPyramid_1838246002694
// MI455X (gfx1250) — compile-verified
//
#include <hip/hip_runtime.h>

typedef __attribute__((ext_vector_type(2))) float v2f;
typedef __attribute__((ext_vector_type(8))) float v8f;

#define NDISP 9   // 2*MAX_DISP+1, MAX_DISP=4

// One K-step (4 channels) of fragments: two A tiles (f1 cols w0, w0+16) and
// three B tiles (f2 cols w0-8, w0+8, w0+24). Loads are unconditional; OOB
// columns use clamped (valid) addresses and are zeroed at compute via masks.
struct Frag { v2f a0, a1, b0, b1, b2; };

__device__ __forceinline__ void load_frag(
    const float* __restrict__ f1, const float* __restrict__ f2,
    unsigned koff, unsigned oA0, unsigned oB0, unsigned oB1, unsigned oB2,
    unsigned HW, Frag& f)
{
  f.a0.x = f1[oA0 + koff];        f.a0.y = f1[oA0 + koff + HW];
  f.a1.x = f1[oA0 + koff + 16u];  f.a1.y = f1[oA0 + koff + 16u + HW];
  f.b0.x = f2[oB0 + koff];        f.b0.y = f2[oB0 + koff + HW];
  f.b1.x = f2[oB1 + koff];        f.b1.y = f2[oB1 + koff + HW];
  f.b2.x = f2[oB2 + koff];        f.b2.y = f2[oB2 + koff + HW];
}

__device__ __forceinline__ void compute_frag(
    const Frag& f, float mB0, float mB2,
    v8f& acc00, v8f& acc01, v8f& acc10, v8f& acc11)
{
  v2f b0m; b0m.x = f.b0.x * mB0; b0m.y = f.b0.y * mB0;
  v2f b2m; b2m.x = f.b2.x * mB2; b2m.y = f.b2.y * mB2;
  acc00 = __builtin_amdgcn_wmma_f32_16x16x4_f32(false, f.a0, false, b0m,  (short)0, acc00, false, false);
  acc01 = __builtin_amdgcn_wmma_f32_16x16x4_f32(false, f.a0, false, f.b1, (short)0, acc01, false, false);
  acc10 = __builtin_amdgcn_wmma_f32_16x16x4_f32(false, f.a1, false, f.b1, (short)0, acc10, false, false);
  acc11 = __builtin_amdgcn_wmma_f32_16x16x4_f32(false, f.a1, false, b2m,  (short)0, acc11, false, false);
}

// One wave (32 lanes) per (b, h, 32-wide w tile-pair).
// acc00 = A0xB0, acc01 = A0xB1, acc10 = A1xB1, acc11 = A1xB2.
// Band extraction: d = j-i-4 (acc00/acc10), d = j-i+12 (acc01/acc11).
__global__ __launch_bounds__(32) void corr_wmma_kernel(
    const float* __restrict__ f1, const float* __restrict__ f2,
    float* __restrict__ out, int C, int H, int W)
{
  const int ntp = W >> 5;                 // 32-column tile pairs per row
  const int bid = blockIdx.x;
  const int tp  = bid % ntp;
  const int h   = (bid / ntp) % H;
  const int bb  = bid / (ntp * H);

  const int lane = threadIdx.x;           // 0..31
  const int li   = lane & 15;
  const int kh   = lane >> 4;             // channel sub-offset 2*kh

  const unsigned HW = (unsigned)(H * W);
  const int w0 = tp << 5;

  // Edge masks (loop-invariant). Middle B tile (w0+8+li) is always in range.
  const int wB0 = w0 - 8 + li;
  const int wB2 = w0 + 24 + li;
  const float mB0 = (wB0 >= 0) ? 1.0f : 0.0f;
  const float mB2 = (wB2 < W) ? 1.0f : 0.0f;
  const int cB0 = (wB0 >= 0) ? wB0 : 0;       // clamped (valid) columns
  const int cB2 = (wB2 < W) ? wB2 : (W - 1);

  const unsigned rowBase = (unsigned)(bb * C) * HW + (unsigned)(h * W);
  const unsigned chOff   = (unsigned)(2 * kh) * HW;
  const unsigned offA0 = rowBase + chOff + (unsigned)(w0 + li);
  const unsigned offB0 = rowBase + chOff + (unsigned)cB0;
  const unsigned offB1 = rowBase + chOff + (unsigned)(w0 + 8 + li);
  const unsigned offB2 = rowBase + chOff + (unsigned)cB2;
  const unsigned sK = 4u * HW;            // elements per K-step (4 channels)

  v8f z = {0.f, 0.f, 0.f, 0.f, 0.f, 0.f, 0.f, 0.f};
  v8f acc00 = z, acc01 = z, acc10 = z, acc11 = z;

  // Software pipeline, prefetch distance 2: while compute(X) waits on its
  // loads, the other buffer's 10 loads remain in flight.
  Frag fA, fB;
  load_frag(f1, f2, 0u, offA0, offB0, offB1, offB2, HW, fA);
  load_frag(f1, f2, sK, offA0, offB0, offB1, offB2, HW, fB);
  unsigned koff = 2u * sK;

  const int ksteps = C >> 2;              // 4 channels per WMMA (C multiple of 8)
  for (int k = 0; k < ksteps - 2; k += 2) {
    compute_frag(fA, mB0, mB2, acc00, acc01, acc10, acc11);
    load_frag(f1, f2, koff, offA0, offB0, offB1, offB2, HW, fA);
    koff += sK;
    compute_frag(fB, mB0, mB2, acc00, acc01, acc10, acc11);
    load_frag(f1, f2, koff, offA0, offB0, offB1, offB2, HW, fB);
    koff += sK;
  }
  compute_frag(fA, mB0, mB2, acc00, acc01, acc10, acc11);
  compute_frag(fB, mB0, mB2, acc00, acc01, acc10, acc11);

  // D layout: lane holds D[v + 8*kh][li] in acc[v].
  const float scale = 1.0f / (float)C;
  const int j = li;
  #pragma unroll
  for (int v = 0; v < 8; ++v) {
    const int i  = v + 8 * kh;
    const int d0 = j - i - 4;   // from B tile left of each A tile
    const int d1 = j - i + 12;  // from B tile right of each A tile
    if (d0 >= 0 && d0 < NDISP) {
      const size_t o = (((size_t)bb * NDISP + d0) * H + h) * W + w0 + i;
      out[o]      = acc00[v] * scale;   // w = w0 + i
      out[o + 16] = acc10[v] * scale;   // w = w0 + 16 + i
    }
    if (d1 >= 0 && d1 < NDISP) {
      const size_t o = (((size_t)bb * NDISP + d1) * H + h) * W + w0 + i;
      out[o]      = acc01[v] * scale;
      out[o + 16] = acc11[v] * scale;
    }
  }
}

// 2x2 average pool, stride 2. H, W are INPUT dims; total = #output elements.
__global__ void avgpool2_kernel(const float* __restrict__ in,
                                float* __restrict__ out,
                                int H, int W, int total)
{
  int idx = blockIdx.x * blockDim.x + threadIdx.x;
  if (idx >= total) return;
  const int Wo = W >> 1, Ho = H >> 1;
  const int wo = idx % Wo;
  const int ho = (idx / Wo) % Ho;
  const int c  = idx / (Wo * Ho);   // flattened (b*C + c)
  const float2* p = (const float2*)(in + ((size_t)c * H + 2 * ho) * (size_t)W) + wo;
  const float2 t = p[0];            // row 2*ho, cols 2wo, 2wo+1
  const float2 u = p[Wo];           // row 2*ho+1
  out[idx] = 0.25f * ((t.x + t.y) + (u.x + u.y));
}

extern "C" void kernel_launch(void* const* d_in, const int* in_sizes, int n_in,
                              void* d_out, int out_size, void* d_ws, size_t ws_size,
                              hipStream_t stream) {
  (void)in_sizes; (void)n_in; (void)out_size; (void)ws_size;
  const float* f1 = (const float*)d_in[0];
  const float* f2 = (const float*)d_in[1];
  float* out = (float*)d_out;

  const int B = 8, C = 256, H0 = 96, W0 = 128;
  const int H1 = H0 / 2, W1 = W0 / 2;
  const int H2 = H0 / 4, W2 = W0 / 4;

  // Workspace: pooled feature maps (fp32), ~60 MB total.
  float* ws    = (float*)d_ws;
  float* f1_l1 = ws;
  float* f2_l1 = f1_l1 + (size_t)B * C * H1 * W1;
  float* f1_l2 = f2_l1 + (size_t)B * C * H1 * W1;
  float* f2_l2 = f1_l2 + (size_t)B * C * H2 * W2;

  float* out0 = out;
  float* out1 = out0 + (size_t)B * NDISP * H0 * W0;
  float* out2 = out1 + (size_t)B * NDISP * H1 * W1;

  const dim3 wblk(32);

  // Level 0 correlation.
  corr_wmma_kernel<<<B * H0 * (W0 / 32), wblk, 0, stream>>>(f1, f2, out0, C, H0, W0);

  // Pool level 0 -> level 1 (mostly L2 hits: 192 MB global L2).
  const int tot1 = B * C * H1 * W1;
  avgpool2_kernel<<<(tot1 + 255) / 256, 256, 0, stream>>>(f1, f1_l1, H0, W0, tot1);
  avgpool2_kernel<<<(tot1 + 255) / 256, 256, 0, stream>>>(f2, f2_l1, H0, W0, tot1);

  // Level 1 correlation.
  corr_wmma_kernel<<<B * H1 * (W1 / 32), wblk, 0, stream>>>(f1_l1, f2_l1, out1, C, H1, W1);

  // Pool level 1 -> level 2.
  const int tot2 = B * C * H2 * W2;
  avgpool2_kernel<<<(tot2 + 255) / 256, 256, 0, stream>>>(f1_l1, f1_l2, H1, W1, tot2);
  avgpool2_kernel<<<(tot2 + 255) / 256, 256, 0, stream>>>(f2_l1, f2_l2, H1, W1, tot2);

  // Level 2 correlation.
  corr_wmma_kernel<<<B * H2 * (W2 / 32), wblk, 0, stream>>>(f1_l2, f2_l2, out2, C, H2, W2);
}